// Codebook_67242007986786
// MI455X (gfx1250) — compile-verified
//
#include <hip/hip_runtime.h>
#include <hip/hip_bf16.h>
#include <math.h>

#define N_TOKENS 32768
#define EMB_N    2048
#define EMB_D    512
#define BETA     0.25f
#define LDS_PITCH 516                    // 512 + 4 dword pad -> no LDS bank conflicts
#define TILE_FLOATS (16 * LDS_PITCH)     // one 16-row B tile (padded)

typedef __attribute__((ext_vector_type(2))) float v2f;
typedef __attribute__((ext_vector_type(4))) float v4f;
typedef __attribute__((ext_vector_type(8))) float v8f;

// d_ws layout (floats): [0,2048) enorm | [2048,4096) counts | [4096] loss accum

__global__ __launch_bounds__(256) void prep_kernel(const float* __restrict__ emb,
                                                   float* __restrict__ ws) {
    int j = blockIdx.x * blockDim.x + threadIdx.x;   // 8 blocks * 256 = 2048
    if (j < EMB_N) {
        const v4f* row = reinterpret_cast<const v4f*>(emb + (size_t)j * EMB_D);
        float s = 0.f;
        #pragma unroll 4
        for (int k = 0; k < EMB_D / 4; ++k) {
            v4f v = row[k];
            s += v.x * v.x + v.y * v.y + v.z * v.z + v.w * v.w;
        }
        ws[j] = s;            // ||e_j||^2
        ws[2048 + j] = 0.f;   // zero histogram each call (graph-replay safe)
    }
    if (j == 0) ws[4096] = 0.f;
}

// Async prefetch of one 16x512 f32 embedding tile into the LDS buffer `buf`.
// The ptrtoint of &buf[...] both captures the LDS array (so its loads can't be
// folded to undef) and yields the wave-relative LDS byte offset in the low 32
// bits.
// 128 threads x 16 iterations x 16B = 32 KB, tracked by ASYNCcnt.
__device__ __forceinline__ void prefetch_tile(const float* __restrict__ emb,
                                              int jt, float* buf, int tid) {
    #pragma unroll 4
    for (int t = tid; t < 16 * (EMB_D / 4); t += 128) {
        int row = t >> 7;                 // t / 128
        int kk  = (t & 127) << 2;         // dword col
        const float* g = &emb[(size_t)(jt + row) * EMB_D + kk];
        unsigned ldsOff = (unsigned)(unsigned long long)&buf[row * LDS_PITCH + kk];
        asm volatile("global_load_async_to_lds_b128 %0, %1, off"
                     :: "v"(ldsOff), "v"(g) : "memory");
    }
}

// Fused GEMM + argmin. Block = 128 threads = 4 waves; wave w owns rows
// (blockIdx.x*4 + w)*16. Embedding j-tiles double-buffered in LDS via
// async-to-LDS DMA; K split into two independent WMMA accumulator chains.
__global__ __launch_bounds__(128) void argmin_kernel(const float* __restrict__ z,
                                                     const float* __restrict__ emb,
                                                     const float* __restrict__ enorm,
                                                     float* __restrict__ outIdx) {
    __shared__ float Bs[2 * TILE_FLOATS];            // 66 KB double buffer
    const int tid  = threadIdx.x;
    const int wave = tid >> 5;
    const int lane = tid & 31;
    const int lo   = lane & 15;
    const int hi   = lane >> 4;
    const int rowBase = (blockIdx.x * 4 + wave) * 16;
    const float* aRow = z + (size_t)(rowBase + lo) * EMB_D;

    float best[8];
    int   bidx[8];
    #pragma unroll
    for (int r = 0; r < 8; ++r) { best[r] = 3.4e38f; bidx[r] = 0; }

    prefetch_tile(emb, 0, Bs, tid);                  // warm up buffer 0

    int sel = 0;
    for (int jt = 0; jt < EMB_N; jt += 16, sel ^= 1) {
        asm volatile("s_wait_asynccnt 0x0" ::: "memory");  // my share of cur landed
        __syncthreads();                                    // everyone's share landed;
                                                            // everyone done reading nxt
        if (jt + 16 < EMB_N)
            prefetch_tile(emb, jt + 16, Bs + (sel ^ 1) * TILE_FLOATS, tid);

        // two independent accumulation chains over K halves
        v8f acc0 = {}, acc1 = {};
        const float* bRow = Bs + sel * TILE_FLOATS + lo * LDS_PITCH;
        #pragma unroll 4
        for (int k0 = 0; k0 < EMB_D / 2; k0 += 4) {
            // A frag: A[m = lo][k = k0 + 2*hi + v]; B frag: B[k][n = lo]
            v2f a0 = *reinterpret_cast<const v2f*>(aRow + k0 + 2 * hi);
            v2f b0 = *reinterpret_cast<const v2f*>(bRow + k0 + 2 * hi);
            acc0 = __builtin_amdgcn_wmma_f32_16x16x4_f32(
                false, a0, false, b0, (short)0, acc0, false, false);
            v2f a1 = *reinterpret_cast<const v2f*>(aRow + (EMB_D / 2) + k0 + 2 * hi);
            v2f b1 = *reinterpret_cast<const v2f*>(bRow + (EMB_D / 2) + k0 + 2 * hi);
            acc1 = __builtin_amdgcn_wmma_f32_16x16x4_f32(
                false, a1, false, b1, (short)0, acc1, false, false);
        }

        // running argmin: score = ||e_j||^2 - 2 * (z_i . e_j)
        int   j  = jt + lo;
        float en = enorm[j];
        #pragma unroll
        for (int r = 0; r < 8; ++r) {
            float val = en - 2.0f * (acc0[r] + acc1[r]);
            if (val < best[r]) { best[r] = val; bidx[r] = j; }
        }
    }

    // butterfly argmin across the 16 lanes of each half-wave (C/D layout halves)
    #pragma unroll
    for (int r = 0; r < 8; ++r) {
        #pragma unroll
        for (int m = 1; m < 16; m <<= 1) {
            float oV = __shfl_xor(best[r], m, 16);
            int   oI = __shfl_xor(bidx[r], m, 16);
            if (oV < best[r] || (oV == best[r] && oI < bidx[r])) {
                best[r] = oV; bidx[r] = oI;
            }
        }
    }
    if (lo == 0) {
        #pragma unroll
        for (int r = 0; r < 8; ++r)
            outIdx[rowBase + 8 * hi + r] = (float)bidx[r];
    }
}

// One block per token: gather z_q, write z_q_st (== z_q numerically),
// accumulate sum((z_q - z)^2) and the float histogram.
__global__ __launch_bounds__(256) void gather_kernel(const float* __restrict__ z,
                                                     const float* __restrict__ emb,
                                                     const float* __restrict__ idxF,
                                                     float* __restrict__ outZq,
                                                     float* __restrict__ counts,
                                                     float* __restrict__ lossAcc) {
    __shared__ float red[256];
    const int t   = blockIdx.x;
    const int idx = (int)idxF[t];
    const int d0  = threadIdx.x * 2;
    v2f zq = *reinterpret_cast<const v2f*>(emb + (size_t)idx * EMB_D + d0);
    v2f zv = *reinterpret_cast<const v2f*>(z   + (size_t)t   * EMB_D + d0);
    *reinterpret_cast<v2f*>(outZq + (size_t)t * EMB_D + d0) = zq;
    float dx = zq.x - zv.x, dy = zq.y - zv.y;
    red[threadIdx.x] = dx * dx + dy * dy;
    __syncthreads();
    for (int s = 128; s > 0; s >>= 1) {
        if (threadIdx.x < s) red[threadIdx.x] += red[threadIdx.x + s];
        __syncthreads();
    }
    if (threadIdx.x == 0) {
        atomicAdd(lossAcc, red[0]);
        atomicAdd(&counts[idx], 1.0f);
    }
}

__global__ __launch_bounds__(256) void finalize_kernel(const float* __restrict__ ws,
                                                       float* __restrict__ outScalars) {
    __shared__ float red[256];
    float s = 0.f;
    for (int j = threadIdx.x; j < EMB_N; j += 256) {
        float p = ws[2048 + j] * (1.0f / (float)N_TOKENS);
        s += p * logf(p + 1e-10f);
    }
    red[threadIdx.x] = s;
    __syncthreads();
    for (int st = 128; st > 0; st >>= 1) {
        if (threadIdx.x < st) red[threadIdx.x] += red[threadIdx.x + st];
        __syncthreads();
    }
    if (threadIdx.x == 0) {
        outScalars[0] = ws[4096] * (1.0f + BETA) / ((float)N_TOKENS * (float)EMB_D);
        outScalars[1] = expf(-red[0]);
    }
}

extern "C" void kernel_launch(void* const* d_in, const int* in_sizes, int n_in,
                              void* d_out, int out_size, void* d_ws, size_t ws_size,
                              hipStream_t stream) {
    const float* z   = (const float*)d_in[0];   // (32768, 512) f32
    const float* emb = (const float*)d_in[1];   // (2048, 512)  f32
    float* out = (float*)d_out;
    float* ws  = (float*)d_ws;

    float* outZq      = out;                                   // 32768*512
    float* outIdx     = out + (size_t)N_TOKENS * EMB_D;        // 32768
    float* outScalars = outIdx + N_TOKENS;                     // loss, perplexity

    prep_kernel   <<<EMB_N / 256,   256, 0, stream>>>(emb, ws);
    argmin_kernel <<<N_TOKENS / 64, 128, 0, stream>>>(z, emb, ws, outIdx);
    gather_kernel <<<N_TOKENS,      256, 0, stream>>>(z, emb, outIdx, outZq,
                                                      ws + 2048, ws + 4096);
    finalize_kernel<<<1,            256, 0, stream>>>(ws, outScalars);
}